// LossYOLOv3_63436666962561
// MI455X (gfx1250) — compile-verified
//
#include <hip/hip_runtime.h>
#include <math.h>

typedef __attribute__((ext_vector_type(2))) float v2f;
typedef __attribute__((ext_vector_type(8))) float v8f;

#define NANCH 10647
#define NCH   25
#define NCLS  20
#define MAXNB 8

__device__ __forceinline__ float sigm(float x) { return 1.0f / (1.0f + expf(-x)); }

// focal(logits, y) single-element term (alpha=0.25, gamma=2)
__device__ __forceinline__ float focal_term(float x, float y) {
    float bce = fmaxf(x, 0.0f) - x * y + log1pf(expf(-fabsf(x)));
    float pr  = sigm(x);
    float pt  = y * pr + (1.0f - y) * (1.0f - pr);
    float af  = y * 0.25f + (1.0f - y) * 0.75f;
    float om  = 1.0f - pt;
    return bce * af * om * om;
}

// Wave32 cross-lane sum via V_WMMA_F32_16X16X4_F32 with B = ones.
// A[m][0] = s (lanes 0-15), A[m][2] = s (lanes 16-31), K=1,3 zero.
// D[m][n] = s_m + s_{m+16}. Lane 0 holds M=0..7, lane 16 holds M=8..15
// of column 0, so those two lanes' 8-element sums are the two half-totals.
// EXEC must be all-ones at the call site (no divergent callers).
__device__ __forceinline__ void waveReduceAtomic(float s, float* acc) {
    v2f a; a[0] = s;    a[1] = 0.0f;
    v2f b; b[0] = 1.0f; b[1] = 1.0f;
    v8f c = {};
    v8f d = __builtin_amdgcn_wmma_f32_16x16x4_f32(false, a, false, b,
                                                  (short)0, c, false, false);
    float t = d[0] + d[1] + d[2] + d[3] + d[4] + d[5] + d[6] + d[7];
    unsigned l = threadIdx.x & 31u;
    if (l == 0u || l == 16u) atomicAdd(acc, t);
}

// One block per image: anchor matching + cls focal + box IoU losses.
__global__ void __launch_bounds__(192)
yolo_match_cls_box(const float* __restrict__ p,
                   const float* __restrict__ gboxes,
                   const int*   __restrict__ glabels,
                   const float* __restrict__ ancs,
                   const float* __restrict__ fsz,
                   int*   __restrict__ mp_out,
                   float* __restrict__ out,
                   int nb, float inv_scale)
{
    __shared__ float gb[MAXNB][4];
    __shared__ int   gl[MAXNB];
    __shared__ float gx[MAXNB], gy[MAXNB], gw[MAXNB], gh[MAXNB];
    __shared__ float ciou[MAXNB][MAXNB * 9];
    __shared__ int   mp[MAXNB];
    __shared__ float acc;

    const int img = blockIdx.x;
    const int tid = threadIdx.x;
    const float eps = 1e-7f;

    if (tid == 0) acc = 0.0f;
    if (tid < nb * 4) gb[tid >> 2][tid & 3] = gboxes[(size_t)img * nb * 4 + tid];
    if (tid < nb)     gl[tid] = glabels[(size_t)img * nb + tid];
    __syncthreads();

    if (tid < nb) {
        float l = gb[tid][0], t = gb[tid][1], r = gb[tid][2], bt = gb[tid][3];
        gx[tid] = (l + r) * 0.5f; gy[tid] = (t + bt) * 0.5f;
        gw[tid] = r - l;          gh[tid] = bt - t;
    }
    __syncthreads();

    const float fs9[9]  = {13.f,13.f,13.f,26.f,26.f,26.f,52.f,52.f,52.f};
    const float ancw[9] = {116.f/416.f,156.f/416.f,373.f/416.f,
                           30.f/416.f, 62.f/416.f, 59.f/416.f,
                           10.f/416.f, 16.f/416.f, 33.f/416.f};
    const float anch[9] = { 90.f/416.f,198.f/416.f,326.f/416.f,
                            61.f/416.f, 45.f/416.f,119.f/416.f,
                            13.f/416.f, 30.f/416.f, 23.f/416.f};
    const float kPI2 = (float)(4.0 / (M_PI * M_PI));

    const int ncol = nb * 9;
    if (tid < ncol) {
        int jb = tid / 9, ja = tid % 9;
        float f = fs9[ja];
        float cellx = floorf(gx[jb] * f) / f;
        float celly = floorf(gy[jb] * f) / f;
        float aw = ancw[ja], ah = anch[ja];
        float off = (float)jb;
        float bx0 = cellx - aw * 0.5f + off;
        float by0 = celly - ah * 0.5f + off;
        float bx1 = cellx + aw * 0.5f + off;
        float by1 = celly + ah * 0.5f + off;
        float wbb = bx1 - bx0, hbb = by1 - by0;
        float atb = atanf(wbb / (hbb + eps));
        for (int b = 0; b < nb; ++b) {
            float ob = (float)b;
            float ax0 = gb[b][0] + ob, ay0 = gb[b][1] + ob;
            float ax1 = gb[b][2] + ob, ay1 = gb[b][3] + ob;
            float lt0 = fmaxf(ax0, bx0), lt1 = fmaxf(ay0, by0);
            float rb0 = fminf(ax1, bx1), rb1 = fminf(ay1, by1);
            float iw = fmaxf(rb0 - lt0, 0.f), ih = fmaxf(rb1 - lt1, 0.f);
            float inter = iw * ih;
            float wa = ax1 - ax0, ha = ay1 - ay0;
            float uni = wa * ha + wbb * hbb - inter;
            float iou = inter / (uni + eps);
            float cl0 = fminf(ax0, bx0), cl1 = fminf(ay0, by0);
            float cr0 = fmaxf(ax1, bx1), cr1 = fmaxf(ay1, by1);
            float cw = cr0 - cl0, ch = cr1 - cl1;
            float c2 = cw * cw + ch * ch + eps;
            float dx = (ax0 + ax1) * 0.5f - (bx0 + bx1) * 0.5f;
            float dy = (ay0 + ay1) * 0.5f - (by0 + by1) * 0.5f;
            float rho2 = dx * dx + dy * dy;
            float dd = atanf(wa / (ha + eps)) - atb;
            float v = kPI2 * dd * dd;
            float alpha = v / (1.0f - iou + v + eps);
            ciou[b][tid] = iou - rho2 / c2 - alpha * v;
        }
    }
    __syncthreads();

    if (tid < nb) {
        float best = -INFINITY; int bj = 0;
        for (int j = 0; j < ncol; ++j) {           // first-max wins, like jnp.argmax
            float cv = ciou[tid][j];
            if (cv > best) { best = cv; bj = j; }
        }
        int a = bj % 9;
        int layer = a / 3;
        float f = (layer == 0) ? 13.f : ((layer == 1) ? 26.f : 52.f);
        float layoff = (layer == 0) ? 0.f : ((layer == 1) ? 507.f : 2535.f);
        float col = floorf(gx[tid] * f);
        float row = floorf(gy[tid] * f);
        int pos = (int)(layoff + (row * f + col) * 3.0f + (float)(a % 3));
        mp[tid] = pos;
        mp_out[(size_t)img * nb + tid] = pos;
    }
    __syncthreads();

    float partial = 0.0f;
    if (tid < nb * NCLS) {                          // cls focal (positives)
        int b = tid / NCLS, c = tid % NCLS;
        size_t base = (size_t)img * NANCH * NCH + (size_t)mp[b] * NCH;
        float x = p[base + 5 + c];
        float y = (c == gl[b] - 1) ? 1.0f : 0.0f;
        partial += focal_term(x, y);
    }
    if (tid < nb) {                                 // box loss on matched anchors
        int b = tid, pos = mp[b];
        size_t base = (size_t)img * NANCH * NCH + (size_t)pos * NCH;
        float px = p[base + 0], py = p[base + 1];
        float pw = p[base + 2], ph = p[base + 3];
        float fx = fsz[(size_t)pos * 2 + 0], fy = fsz[(size_t)pos * 2 + 1];
        float ax = ancs[(size_t)pos * 4 + 0], ay = ancs[(size_t)pos * 4 + 1];
        float aw = ancs[(size_t)pos * 4 + 2], ah = ancs[(size_t)pos * 4 + 3];
        float cx = sigm(px) / fx + ax;
        float cy = sigm(py) / fy + ay;
        float ww = expf(pw) * aw;
        float hh = expf(ph) * ah;
        float b0 = cx - ww * 0.5f, b1 = cy - hh * 0.5f;
        float b2 = cx + ww * 0.5f, b3 = cy + hh * 0.5f;
        float a0 = gb[b][0], a1 = gb[b][1], a2 = gb[b][2], a3 = gb[b][3];
        float lt0 = fmaxf(a0, b0), lt1 = fmaxf(a1, b1);
        float rb0 = fminf(a2, b2), rb1 = fminf(a3, b3);
        float iw = fmaxf(rb0 - lt0, 0.f), ih = fmaxf(rb1 - lt1, 0.f);
        float inter = iw * ih;
        float areaA = (a2 - a0) * (a3 - a1);
        float areaB = (b2 - b0) * (b3 - b1);
        float iou = inter / (areaA + areaB - inter + eps);
        float w = 2.0f - gw[b] * gh[b];
        partial += w * (1.0f - iou);
    }
    // EXEC is all-ones here (6 full waves, no early returns) -> WMMA reduce.
    waveReduceAtomic(partial * inv_scale, &acc);
    __syncthreads();
    if (tid == 0) atomicAdd(out, acc);
}

// Conf focal over all anchors. grid = (ceil(10647/256), B)
__global__ void __launch_bounds__(256)
yolo_conf(const float* __restrict__ p,
          const int*   __restrict__ mp_in,
          float* __restrict__ out,
          int nb, float inv_scale)
{
    __shared__ int   mp[MAXNB];
    __shared__ float acc;
    const int img = blockIdx.y;
    const int tid = threadIdx.x;
    if (tid == 0) acc = 0.0f;
    if (tid < nb) mp[tid] = mp_in[(size_t)img * nb + tid];
    __syncthreads();

    int i = blockIdx.x * blockDim.x + tid;
    float s = 0.0f;
    if (i < NANCH) {
        float x = p[(size_t)img * NANCH * NCH + (size_t)i * NCH + 4];
        float y = 0.0f;
        #pragma unroll
        for (int k = 0; k < MAXNB; ++k)
            if (k < nb && i == mp[k]) y = 1.0f;     // scatter-set semantics (dup-safe)
        s = focal_term(x, y) * inv_scale;
    }
    // EXEC all-ones (8 full waves reconverged) -> WMMA wave reduction.
    waveReduceAtomic(s, &acc);
    __syncthreads();
    if (tid == 0) atomicAdd(out, acc);
}

extern "C" void kernel_launch(void* const* d_in, const int* in_sizes, int n_in,
                              void* d_out, int out_size, void* d_ws, size_t ws_size,
                              hipStream_t stream) {
    const float* p       = (const float*)d_in[0];
    const float* gboxes  = (const float*)d_in[1];
    const int*   glabels = (const int*)  d_in[2];
    const float* ancs    = (const float*)d_in[3];
    const float* fsz     = (const float*)d_in[4];

    const int Bimg = in_sizes[0] / (NANCH * NCH);          // 128
    int nb = in_sizes[2] / Bimg;                           // 8
    if (nb > MAXNB) nb = MAXNB;
    const float inv_scale = 1.0f / ((float)nb * (float)Bimg);

    int*   ws_mp = (int*)d_ws;                             // B*nb match positions
    float* out   = (float*)d_out;

    hipMemsetAsync(d_out, 0, (size_t)out_size * sizeof(float), stream);

    yolo_match_cls_box<<<Bimg, 192, 0, stream>>>(p, gboxes, glabels, ancs, fsz,
                                                 ws_mp, out, nb, inv_scale);

    dim3 gridB((NANCH + 255) / 256, Bimg);
    yolo_conf<<<gridB, 256, 0, stream>>>(p, ws_mp, out, nb, inv_scale);
}